// Correlation2D_20109036880099
// MI455X (gfx1250) — compile-verified
//
#include <hip/hip_runtime.h>

// Correlation2D (FlowNet-style), PAD=3 K=3 MD=1 S1=S2=1, fp32, gfx1250.
// Bandwidth-bound (~211 MB -> ~9us floor @ 23.3 TB/s); channel reduction is
// mapped onto V_WMMA_F32_16X16X4_F32 (full fp32, compute is ~1000x below
// roofline so WMMA waste is free). Diagonal-band trick: A = 16 x1 pixels of a
// row, B = 16 x2 pixels offset by -1; D[m, m+1+dx] is the dx in {-1,0,1}
// correlation for pixel m. Three accumulator chains (one per dy) share A.
// Interior tiles (~81%) run an unguarded straight-line loop; edge tiles use
// clamped addresses + v_cndmask (no exec-mask branching). Depth-1 software
// pipeline overlaps the next K-step's 8 loads with the current 3 WMMAs.

typedef __attribute__((ext_vector_type(2))) float v2f;
typedef __attribute__((ext_vector_type(8))) float v8f;

#define CC 96
#define HH 256
#define WW 256
#define OH 258
#define OW 258
#define TH 14              // output rows per tile
#define TW 12              // output cols per tile
#define QR (TH + 2)        // q rows per tile (16)
#define QC (TW + 2)        // q cols per tile (14)
#define ND 9
#define KSTEPS (CC / 4)    // 24 WMMA K-steps per chain
static constexpr long HW = (long)HH * WW;

__device__ __forceinline__ int iclamp(int v, int lo, int hiv) {
    return v < lo ? lo : (v > hiv ? hiv : v);
}

struct Bases {
    long a, b0, b1, b2;          // per-lane element bases (hi folded in)
    bool aok, b0ok, b1ok, b2ok;  // only used on EDGE path
};

template<bool EDGE>
__device__ __forceinline__ void load_step(const float* __restrict__ x1,
                                          const float* __restrict__ x2,
                                          const Bases& o, long cOff,
                                          v2f& a, v2f& b0, v2f& b1, v2f& b2)
{
    // a.{x,y} = channels c0+2hi, c0+2hi+1 ; b.{x,y} = channels c0+hi, c0+hi+2
    float a0 = x1[o.a  + cOff];
    float a1 = x1[o.a  + cOff + HW];
    float p0 = x2[o.b0 + cOff];
    float p1 = x2[o.b0 + cOff + 2 * HW];
    float q0 = x2[o.b1 + cOff];
    float q1 = x2[o.b1 + cOff + 2 * HW];
    float r0 = x2[o.b2 + cOff];
    float r1 = x2[o.b2 + cOff + 2 * HW];
    if (EDGE) {   // addresses are clamped-valid; zero OOB lanes via cndmask
        a0 = o.aok  ? a0 : 0.0f;  a1 = o.aok  ? a1 : 0.0f;
        p0 = o.b0ok ? p0 : 0.0f;  p1 = o.b0ok ? p1 : 0.0f;
        q0 = o.b1ok ? q0 : 0.0f;  q1 = o.b1ok ? q1 : 0.0f;
        r0 = o.b2ok ? r0 : 0.0f;  r1 = o.b2ok ? r1 : 0.0f;
    }
    a.x = a0;  a.y = a1;
    b0.x = p0; b0.y = p1;
    b1.x = q0; b1.y = q1;
    b2.x = r0; b2.y = r1;
}

template<bool EDGE>
__device__ __forceinline__ void corr_tile(const float* __restrict__ x1,
                                          const float* __restrict__ x2,
                                          float* __restrict__ out,
                                          float (*dtile)[256],
                                          float (*qbuf)[ND][QC],
                                          int b, int i0, int j0)
{
    const int lane = threadIdx.x;     // 0..31 (wave32)
    const int m    = lane & 15;
    const int hi   = lane >> 4;

    const long batchOff = (long)b * CC * HW;
    const int colA = j0 - 2 + m;      // x1 col
    const int colB = j0 - 3 + m;      // x2 col

    for (int rho = 0; rho < QR; ++rho) {
        const int u = i0 - 2 + rho;   // q row == x1 row
        Bases o;
        o.aok = o.b0ok = o.b1ok = o.b2ok = true;
        if (!EDGE) {
            o.a  = batchOff + (long)u * WW       + colA + (long)(2 * hi) * HW;
            o.b0 = batchOff + (long)(u - 1) * WW + colB + (long)hi * HW;
            o.b1 = batchOff + (long)u * WW       + colB + (long)hi * HW;
            o.b2 = batchOff + (long)(u + 1) * WW + colB + (long)hi * HW;
        } else {
            const bool colAok = (colA >= 0) & (colA < WW);
            const bool colBok = (colB >= 0) & (colB < WW);
            const int cA = iclamp(colA, 0, WW - 1);
            const int cB = iclamp(colB, 0, WW - 1);
            const int u0 = iclamp(u - 1, 0, HH - 1);
            const int u1 = iclamp(u,     0, HH - 1);
            const int u2 = iclamp(u + 1, 0, HH - 1);
            o.aok  = colAok & (u >= 0) & (u < HH);
            o.b0ok = colBok & (u - 1 >= 0) & (u - 1 < HH);
            o.b1ok = colBok & (u >= 0) & (u < HH);
            o.b2ok = colBok & (u + 1 >= 0) & (u + 1 < HH);
            o.a  = batchOff + (long)u1 * WW + cA + (long)(2 * hi) * HW;
            o.b0 = batchOff + (long)u0 * WW + cB + (long)hi * HW;
            o.b1 = batchOff + (long)u1 * WW + cB + (long)hi * HW;
            o.b2 = batchOff + (long)u2 * WW + cB + (long)hi * HW;
        }

        v8f acc0 = {};   // dy = -1
        v8f acc1 = {};   // dy =  0
        v8f acc2 = {};   // dy = +1
        v2f a, b0v, b1v, b2v;
        load_step<EDGE>(x1, x2, o, 0, a, b0v, b1v, b2v);

#pragma unroll 4
        for (int kk = 0; kk < KSTEPS - 1; ++kk) {
            v2f an, b0n, b1n, b2n;   // prefetch next K-step while WMMAs run
            load_step<EDGE>(x1, x2, o, (long)(kk + 1) * 4 * HW, an, b0n, b1n, b2n);
            acc0 = __builtin_amdgcn_wmma_f32_16x16x4_f32(
                       false, a, false, b0v, (short)0, acc0, false, false);
            acc1 = __builtin_amdgcn_wmma_f32_16x16x4_f32(
                       false, a, false, b1v, (short)0, acc1, false, false);
            acc2 = __builtin_amdgcn_wmma_f32_16x16x4_f32(
                       false, a, false, b2v, (short)0, acc2, false, false);
            a = an; b0v = b0n; b1v = b1n; b2v = b2n;
        }
        acc0 = __builtin_amdgcn_wmma_f32_16x16x4_f32(
                   false, a, false, b0v, (short)0, acc0, false, false);
        acc1 = __builtin_amdgcn_wmma_f32_16x16x4_f32(
                   false, a, false, b1v, (short)0, acc1, false, false);
        acc2 = __builtin_amdgcn_wmma_f32_16x16x4_f32(
                   false, a, false, b2v, (short)0, acc2, false, false);

        // Dump all three D tiles, one barrier, extract band, one barrier.
        const int M = 8 * hi;             // C/D layout: lanes16-31 -> M+8
#pragma unroll
        for (int g = 0; g < 8; ++g) {
            dtile[0][(M + g) * 16 + m] = acc0[g];
            dtile[1][(M + g) * 16 + m] = acc1[g];
            dtile[2][(M + g) * 16 + m] = acc2[g];
        }
        __syncthreads();
        if (lane < QC) {
#pragma unroll
            for (int d = 0; d < 3; ++d) {
                // dx = -1, 0, +1  ->  n = m, m+1, m+2
                qbuf[rho][d * 3 + 0][lane] = dtile[d][lane * 16 + lane];
                qbuf[rho][d * 3 + 1][lane] = dtile[d][lane * 16 + lane + 1];
                qbuf[rho][d * 3 + 2][lane] = dtile[d][lane * 16 + lane + 2];
            }
        }
        __syncthreads();
    }

    // 3x3 box sum + normalization epilogue.
    const float norm = 1.0f / (float)(9 * CC);
    for (int idx = lane; idx < ND * TH * TW; idx += 32) {
        const int dd = idx / (TH * TW);
        const int r  = idx % (TH * TW);
        const int s  = r / TW;
        const int t  = r % TW;
        const int i  = i0 + s;
        const int j  = j0 + t;
        if ((i < OH) & (j < OW)) {
            float sum = 0.0f;
#pragma unroll
            for (int aa = 0; aa < 3; ++aa)
#pragma unroll
                for (int bb = 0; bb < 3; ++bb)
                    sum += qbuf[s + aa][dd][t + bb];
            out[(((long)b * ND + dd) * OH + i) * OW + j] = sum * norm;
        }
    }
}

__global__ __launch_bounds__(32)
void corr2d_wmma_f32(const float* __restrict__ x1,
                     const float* __restrict__ x2,
                     float* __restrict__ out)
{
    __shared__ float dtile[3][256];
    __shared__ float qbuf[QR][ND][QC];

    const int j0 = blockIdx.x * TW;
    const int i0 = blockIdx.y * TH;
    const int b  = blockIdx.z;

    // Interior iff every address this tile touches is in-bounds:
    // x1 cols [j0-2, j0+13], x2 cols [j0-3, j0+12], x2 rows [i0-3, i0+14].
    const bool interior = (j0 >= 3) && (j0 + 13 <= WW - 1) &&
                          (i0 >= 3) && (i0 + 14 <= HH - 1);
    if (interior)
        corr_tile<false>(x1, x2, out, dtile, qbuf, b, i0, j0);
    else
        corr_tile<true>(x1, x2, out, dtile, qbuf, b, i0, j0);
}

extern "C" void kernel_launch(void* const* d_in, const int* in_sizes, int n_in,
                              void* d_out, int out_size, void* d_ws, size_t ws_size,
                              hipStream_t stream) {
    const float* x1 = (const float*)d_in[0];
    const float* x2 = (const float*)d_in[1];
    float* out = (float*)d_out;
    const int B = in_sizes[0] / (CC * HH * WW);

    dim3 grid((OW + TW - 1) / TW, (OH + TH - 1) / TH, B);
    corr2d_wmma_f32<<<grid, 32, 0, stream>>>(x1, x2, out);
}